// CLF_QP_Net_24343874634470
// MI455X (gfx1250) — compile-verified
//
#include <hip/hip_runtime.h>
#include <hip/hip_bf16.h>
#include <math.h>

typedef float v2f __attribute__((ext_vector_type(2)));
typedef float v4f __attribute__((ext_vector_type(4)));
typedef float v8f __attribute__((ext_vector_type(8)));

#define H 256
#define H4 64
#define WAVES 4
// per-wave LDS floats: A1(16x256)=4096, A2(16x256)=4096, A3(16x64)=1024,
// gradV(16x4)=64, u(16)=16  -> 9296 floats
#define WSLOT 9296

__device__ __forceinline__ v8f wmma4(v2f a, v2f b, v8f c) {
  // D = A(16x4) * B(4x16) + C, f32. 8 args: neg_a, A, neg_b, B, c_mod, C, reuse_a, reuse_b
  return __builtin_amdgcn_wmma_f32_16x16x4_f32(false, a, false, b, (short)0, c,
                                               false, false);
}

// Branch-free tanh: 1 - 2/(exp(2x)+1). v_exp_f32 + v_rcp_f32, no EXEC writes,
// saturates correctly to +/-1 for large |x|.
__device__ __forceinline__ float tanh_fast(float x) {
  const float e = __expf(2.0f * x);
  return 1.0f - 2.0f * __builtin_amdgcn_rcpf(e + 1.0f);
}

// out_tile(16x16) of act(16xK) @ W^T where W is (Nout,K) row-major, tile j.
template <int K>
__device__ __forceinline__ v8f tile_lds_wT(const float* act, const float* W, int j) {
  const int m  = threadIdx.x & 15;
  const int kh = (threadIdx.x >> 4) & 1;
  const float* a0 = act + m * K + 2 * kh;
  const float* b0 = W + (16 * j + m) * K + 2 * kh;
  v8f c = {};
#pragma unroll 8
  for (int kb = 0; kb < K / 4; ++kb)
    c = wmma4(*(const v2f*)(a0 + 4 * kb), *(const v2f*)(b0 + 4 * kb), c);
  return c;
}

// out_tile(16x16) of act(16xK) @ W where W is (K,N) row-major, tile j.
template <int K, int N>
__device__ __forceinline__ v8f tile_lds_w(const float* act, const float* W, int j) {
  const int n  = threadIdx.x & 15;
  const int kh = (threadIdx.x >> 4) & 1;
  const float* a0 = act + n * K + 2 * kh;  // lane&15 is also A's row m
  v8f c = {};
#pragma unroll 8
  for (int kb = 0; kb < K / 4; ++kb) {
    const int k0 = 4 * kb + 2 * kh;
    v2f b;
    b.x = W[(k0 + 0) * N + 16 * j + n];
    b.y = W[(k0 + 1) * N + 16 * j + n];
    c = wmma4(*(const v2f*)(a0 + 4 * kb), b, c);
  }
  return c;
}

__device__ __forceinline__ v8f bias_tanh(v8f c, const float* bvec, int j) {
  const float b = bvec[16 * j + (threadIdx.x & 15)];
  v8f r;
#pragma unroll
  for (int i = 0; i < 8; ++i) r[i] = tanh_fast(c[i] + b);
  return r;
}

__device__ __forceinline__ void store_tile(float* dst, int K, int j, v8f c) {
  const int n  = threadIdx.x & 15;
  const int mh = ((threadIdx.x >> 4) & 1) * 8;
#pragma unroll
  for (int r = 0; r < 8; ++r) dst[(mh + r) * K + 16 * j + n] = c[r];
}

__device__ __forceinline__ v8f mul_onem2(v8f c, const float* act, int K, int j) {
  const int n  = threadIdx.x & 15;
  const int mh = ((threadIdx.x >> 4) & 1) * 8;
#pragma unroll
  for (int r = 0; r < 8; ++r) {
    const float a = act[(mh + r) * K + 16 * j + n];
    c[r] *= (1.0f - a * a);
  }
  return c;
}

__global__ void __launch_bounds__(WAVES * 32)
clf_qp_net_kernel(const float* __restrict__ x,
                  const float* __restrict__ Vw1, const float* __restrict__ Vb1,
                  const float* __restrict__ Vw2, const float* __restrict__ Vb2,
                  const float* __restrict__ Vw3, const float* __restrict__ Vb3,
                  const float* __restrict__ Uw1, const float* __restrict__ Ub1,
                  const float* __restrict__ Uw2, const float* __restrict__ Ub2,
                  const float* __restrict__ Uw3, const float* __restrict__ Ub3,
                  float* __restrict__ out, int bs) {
  extern __shared__ float smem[];
  const int wave = threadIdx.x >> 5;
  float* A1   = smem + wave * WSLOT;  // 16x256: a1, later t1, later u2
  float* A2   = A1 + 4096;            // 16x256: a2, later t2, later u1
  float* A3   = A2 + 4096;            // 16x64 : a3
  float* ROWG = A3 + 1024;            // 16x4  : grad_V
  float* ROWU = ROWG + 64;            // 16    : u

  const int lane = threadIdx.x & 31;
  const int m    = lane & 15;
  const int kh   = (lane >> 4) & 1;
  const int row0 = (blockIdx.x * WAVES + wave) * 16;

  // warm L2 for the large weight matrices (they stay resident)
  __builtin_prefetch(Vw2 + (threadIdx.x << 6), 0, 1);
  __builtin_prefetch(Uw2 + (threadIdx.x << 6), 0, 1);

  // A operand for layer 1: x tile (16x4), built straight from global memory.
  const v2f ax = *(const v2f*)(x + (row0 + m) * 4 + 2 * kh);

  // ---- V forward ----
  // a1 = tanh(x @ Vw1^T + Vb1)      (single K=4 WMMA per tile)
  for (int j = 0; j < 16; ++j) {
    v8f cz = {};
    v8f c = wmma4(ax, *(const v2f*)(Vw1 + (16 * j + m) * 4 + 2 * kh), cz);
    store_tile(A1, H, j, bias_tanh(c, Vb1, j));
  }
  // a2 = tanh(a1 @ Vw2^T + Vb2)
  for (int j = 0; j < 16; ++j)
    store_tile(A2, H, j, bias_tanh(tile_lds_wT<H>(A1, Vw2, j), Vb2, j));
  // a3 = tanh(a2 @ Vw3^T + Vb3)     (16x64)
  for (int j = 0; j < 4; ++j)
    store_tile(A3, H4, j, bias_tanh(tile_lds_wT<H>(A2, Vw3, j), Vb3, j));

  // ---- V backward (vector-Jacobian product) ----
  // t2 = ((a3*(1-a3^2)) @ Vw3) * (1-a2^2) ; overwrite A2
  for (int j = 0; j < 16; ++j) {
    v8f c = {};
#pragma unroll 8
    for (int kb = 0; kb < 16; ++kb) {
      const int k0 = 4 * kb + 2 * kh;
      v2f a = *(const v2f*)(A3 + m * H4 + 4 * kb + 2 * kh);
      a.x = a.x - a.x * a.x * a.x;  // a*(1-a^2)
      a.y = a.y - a.y * a.y * a.y;
      v2f b;
      b.x = Vw3[(k0 + 0) * H + 16 * j + m];
      b.y = Vw3[(k0 + 1) * H + 16 * j + m];
      c = wmma4(a, b, c);
    }
    store_tile(A2, H, j, mul_onem2(c, A2, H, j));
  }
  // t1 = (t2 @ Vw2) * (1-a1^2) ; overwrite A1
  for (int j = 0; j < 16; ++j)
    store_tile(A1, H, j, mul_onem2(tile_lds_w<H, H>(A2, Vw2, j), A1, H, j));
  // grad_V = t1 @ Vw1  (16x4, N-masked single tile, K=256)
  {
    const float* a0 = A1 + m * H + 2 * kh;
    v8f c = {};
#pragma unroll 8
    for (int kb = 0; kb < 64; ++kb) {
      const int k0 = 4 * kb + 2 * kh;
      v2f b;
      if (m < 4) { b.x = Vw1[(k0 + 0) * 4 + m]; b.y = Vw1[(k0 + 1) * 4 + m]; }
      else       { b.x = 0.0f; b.y = 0.0f; }
      c = wmma4(*(const v2f*)(a0 + 4 * kb), b, c);
    }
    const int mh = kh * 8;
    if (m < 4) {
#pragma unroll
      for (int r = 0; r < 8; ++r) ROWG[(mh + r) * 4 + m] = c[r];
    }
  }

  // ---- U forward (reuse buffers) ----
  for (int j = 0; j < 16; ++j) {
    v8f cz = {};
    v8f c = wmma4(ax, *(const v2f*)(Uw1 + (16 * j + m) * 4 + 2 * kh), cz);
    store_tile(A2, H, j, bias_tanh(c, Ub1, j));  // u1 -> A2
  }
  for (int j = 0; j < 16; ++j)
    store_tile(A1, H, j, bias_tanh(tile_lds_wT<H>(A2, Uw2, j), Ub2, j));  // u2 -> A1
  // u = tanh(u2 @ Uw3^T + Ub3) * 20   (N=1, masked tile)
  {
    const float* a0 = A1 + m * H + 2 * kh;
    v8f c = {};
#pragma unroll 8
    for (int kb = 0; kb < 64; ++kb) {
      v2f b;
      if (m == 0) b = *(const v2f*)(Uw3 + 4 * kb + 2 * kh);
      else        { b.x = 0.0f; b.y = 0.0f; }
      c = wmma4(*(const v2f*)(a0 + 4 * kb), b, c);
    }
    const float ub = Ub3[0];
    const int mh = kh * 8;
    if (m == 0) {
#pragma unroll
      for (int r = 0; r < 8; ++r) ROWU[mh + r] = 20.0f * tanh_fast(c[r] + ub);
    }
  }

  // ---- per-row epilogue: V, dynamics, Vdot ----
  if (lane < 16) {
    const int grow = row0 + lane;
    const v4f xr = *(const v4f*)(x + grow * 4);
    const float th = xr[1], vv = xr[2], om = xr[3];
    const float cth = __cosf(th), sth = __sinf(th);
    const float f2 = (cth * (9.8f * sth + 11.5f * vv) + 68.4f * vv -
                      1.2f * om * om * sth) / (cth - 24.7f);
    const float f3 = (-58.8f * vv * cth - 243.5f * vv -
                      sth * (208.3f + om * om * cth)) / (cth * cth - 24.7f);
    const float g2 = (-1.8f * cth - 10.9f) / (cth - 24.7f);
    const float g3 = (9.3f * cth + 38.6f) / (cth * cth - 24.7f);

    const float gv0 = ROWG[lane * 4 + 0], gv1 = ROWG[lane * 4 + 1];
    const float gv2 = ROWG[lane * 4 + 2], gv3 = ROWG[lane * 4 + 3];
    const float u = ROWU[lane];

    float vs = 0.0f;
#pragma unroll 4
    for (int k = 0; k < 16; ++k) {
      const v4f a = *(const v4f*)(A3 + lane * H4 + 4 * k);
      vs += a[0] * a[0] + a[1] * a[1] + a[2] * a[2] + a[3] * a[3];
    }
    const float LfV = gv0 * vv + gv1 * om + gv2 * f2 + gv3 * f3;
    const float LgV = gv2 * g2 + gv3 * g3;

    out[grow]          = u;               // u (BS,1)
    out[bs + grow]     = 0.5f * vs;       // V (BS,)
    out[2 * bs + grow] = LfV + LgV * u;   // Vdot (BS,1,1)
  }
}

extern "C" void kernel_launch(void* const* d_in, const int* in_sizes, int n_in,
                              void* d_out, int out_size, void* d_ws, size_t ws_size,
                              hipStream_t stream) {
  const float* x   = (const float*)d_in[0];
  const float* Vw1 = (const float*)d_in[1];
  const float* Vb1 = (const float*)d_in[2];
  const float* Vw2 = (const float*)d_in[3];
  const float* Vb2 = (const float*)d_in[4];
  const float* Vw3 = (const float*)d_in[5];
  const float* Vb3 = (const float*)d_in[6];
  const float* Uw1 = (const float*)d_in[7];
  const float* Ub1 = (const float*)d_in[8];
  const float* Uw2 = (const float*)d_in[9];
  const float* Ub2 = (const float*)d_in[10];
  const float* Uw3 = (const float*)d_in[11];
  const float* Ub3 = (const float*)d_in[12];
  float* out = (float*)d_out;

  const int bs = in_sizes[0] / 4;              // NX = 4
  const size_t shmem = (size_t)WAVES * WSLOT * sizeof(float);  // ~145 KB / block
  (void)hipFuncSetAttribute((const void*)clf_qp_net_kernel,
                            hipFuncAttributeMaxDynamicSharedMemorySize, (int)shmem);
  dim3 block(WAVES * 32);
  dim3 grid(bs / (WAVES * 16));
  clf_qp_net_kernel<<<grid, block, shmem, stream>>>(
      x, Vw1, Vb1, Vw2, Vb2, Vw3, Vb3, Uw1, Ub1, Uw2, Ub2, Uw3, Ub3, out, bs);
}